// SO3Pooling_55379308314785
// MI455X (gfx1250) — compile-verified
//
#include <hip/hip_runtime.h>

// ---------------- problem constants ----------------
#define NSPEC 5456          // B_OUT*(4*B_OUT^2-1)/3
#define PI_D  3.14159265358979323846264338327950288

typedef __attribute__((ext_vector_type(2))) float v2f;
typedef __attribute__((ext_vector_type(8))) float v8f;

__device__ __forceinline__ int l_off(int l) { return l * (4 * l * l - 1) / 3; }

// ---------------- double-precision helpers for constant generation ----------------
__device__ double dfact(int n) { double f = 1.0; for (int i = 2; i <= n; ++i) f *= (double)i; return f; }
__device__ double dipow(double x, int n) { double r = 1.0; for (int i = 0; i < n; ++i) r *= x; return r; }

__device__ double wig_d(int l, int mp, int m, double cb, double sb) {
  double pref = sqrt(dfact(l + mp) * dfact(l - mp) * dfact(l + m) * dfact(l - m));
  int k0 = (m - mp > 0) ? (m - mp) : 0;
  int k1 = (l - mp < l + m) ? (l - mp) : (l + m);
  double acc = 0.0;
  for (int k = k0; k <= k1; ++k) {
    double den = dfact(l + m - k) * dfact(k) * dfact(l - mp - k) * dfact(mp - m + k);
    double sgn = ((mp - m + k) & 1) ? -1.0 : 1.0;
    acc += sgn / den * dipow(cb, 2 * l + m - mp - 2 * k) * dipow(sb, mp - m + 2 * k);
  }
  return pref * acc;
}

__device__ double squad_w(int bg, int j) {
  double s = 0.0;
  for (int k = 0; k < bg; ++k)
    s += sin((2 * j + 1) * (2 * k + 1) * PI_D / (4.0 * bg)) / (double)(2 * k + 1);
  return 2.0 / bg * sin(PI_D * (2 * j + 1) / (4.0 * bg)) * s;
}

// ---------------- constant-generation kernels ----------------
// Wigner stack: out[j*NSPEC + off_l + e], e = (mp+l)*L + (m+l)
__global__ void k_wigner(float* out, int bg, int weighted) {
  int j = blockIdx.x;   // beta sample, 0..2*bg-1
  int l = blockIdx.y;   // 0..15
  int L = 2 * l + 1;
  double beta = PI_D * (2 * j + 1) / (4.0 * bg);
  double cb = cos(0.5 * beta), sb = sin(0.5 * beta);
  double scale = weighted ? squad_w(bg, j) : (double)L;
  int off = l_off(l);
  for (int e = threadIdx.x; e < L * L; e += blockDim.x) {
    int mp = e / L - l, m = e % L - l;
    out[(size_t)j * NSPEC + off + e] = (float)(wig_d(l, mp, m, cb, sb) * scale);
  }
}

// Grid Wigner-D, stored s-major for the WMMA y-GEMM: D_t[s*72 + g]
__global__ void k_dgrid(float* Dre_t, float* Dim_t) {
  int g = blockIdx.x;   // ib*9 + ia*3 + ic, 0..71
  int l = blockIdx.y;
  int L = 2 * l + 1;
  int ib = g / 9, ia = (g % 9) / 3, ic = g % 3;
  double beta  = (double)(ib + 1) * (PI_D / 8.0) / 3.0;
  double alpha = (double)ia * (2.0 * PI_D / 8.0);
  double gamma = (double)ic * (2.0 * PI_D / 3.0);
  double cb = cos(0.5 * beta), sb = sin(0.5 * beta);
  int off = l_off(l);
  for (int e = threadIdx.x; e < L * L; e += blockDim.x) {
    int mp = e / L - l, m = e % L - l;
    double d  = wig_d(l, mp, m, cb, sb);
    double ph = -((double)mp * alpha + (double)m * gamma);
    Dre_t[(size_t)(off + e) * 72 + g] = (float)(d * cos(ph));
    Dim_t[(size_t)(off + e) * 72 + g] = (float)(d * sin(ph));
  }
}

// spectral index maps (mod 64 for the input grid)
__global__ void k_indices(int* m_in, int* k_in) {
  int l = blockIdx.x;
  int L = 2 * l + 1;
  int off = l_off(l);
  for (int e = threadIdx.x; e < L * L; e += blockDim.x) {
    int m = e / L - l, k = e % L - l;
    m_in[off + e] = (m + 64) & 63;
    k_in[off + e] = (k + 64) & 63;
  }
}

// transpose kernel weights to g-major: kern_t[g*2048 + (i*64+o)]
__global__ void k_kern_t(const float* __restrict__ kern, float* __restrict__ kern_t) {
  int t = blockIdx.x * blockDim.x + threadIdx.x;
  if (t >= 2048 * 72) return;
  int r = t / 72, g = t % 72;
  kern_t[(size_t)g * 2048 + r] = kern[t];
}

// ---------------- LDS batched 1D-FFT pass (N lines of N-point FFT) ----------------
template <int N>
__device__ void fft_pass(float* re, float* im, int nth, int tid, bool colpass, float sign) {
  constexpr int P = N + 1;                 // pad to dodge bank conflicts
  constexpr int LOG2N = (N == 64) ? 6 : 5;
  for (int idx = tid; idx < N * N; idx += nth) {
    int line = idx / N, e = idx % N;
    int r = 0;
#pragma unroll
    for (int bi = 0; bi < LOG2N; ++bi) r |= ((e >> bi) & 1) << (LOG2N - 1 - bi);
    if (e < r) {
      int i0 = colpass ? e * P + line : line * P + e;
      int i1 = colpass ? r * P + line : line * P + r;
      float tr = re[i0]; re[i0] = re[i1]; re[i1] = tr;
      float ti = im[i0]; im[i0] = im[i1]; im[i1] = ti;
    }
  }
  __syncthreads();
  for (int len = 2; len <= N; len <<= 1) {
    int half = len >> 1;
    for (int idx = tid; idx < N * (N / 2); idx += nth) {
      int line = idx / (N / 2), tt = idx % (N / 2);
      int blk = tt / half, pos = tt % half;
      int e0 = blk * len + pos, e1 = e0 + half;
      float ang = sign * 6.283185307179586f * (float)pos / (float)len;
      float wr = __cosf(ang), wim = __sinf(ang);
      int i0 = colpass ? e0 * P + line : line * P + e0;
      int i1 = colpass ? e1 * P + line : line * P + e1;
      float xr = re[i1], xi = im[i1];
      float tr = wr * xr - wim * xi;
      float ti = wr * xi + wim * xr;
      re[i1] = re[i0] - tr; im[i1] = im[i0] - ti;
      re[i0] += tr;         im[i0] += ti;
    }
    __syncthreads();
  }
}

// ---------------- forward FFT (64x64) + compact spectral extraction ----------------
// plane p = (Bb*32+Fi)*64 + b ; writes q-major Xc[q*8192 + p] for 31x31 needed freqs
__global__ void k_fft_fwd(const float* __restrict__ x,
                          float* __restrict__ Xcre, float* __restrict__ Xcim) {
  __shared__ float lre[64 * 65];
  __shared__ float lim[64 * 65];
  int p = blockIdx.x;
  int tid = threadIdx.x;
  size_t base = (size_t)p * 4096;
  for (int idx = tid; idx < 4096; idx += 256) {
    int u = idx >> 6, v = idx & 63;
    lre[u * 65 + v] = x[base + idx];
    lim[u * 65 + v] = 0.f;
  }
  __syncthreads();
  fft_pass<64>(lre, lim, 256, tid, false, -1.f);
  fft_pass<64>(lre, lim, 256, tid, true, -1.f);
  for (int q = tid; q < 961; q += 256) {
    int mi = q / 31, ki = q % 31;
    int m = (mi <= 15) ? mi : mi + 33;   // compact -> mod-64 freq
    int k = (ki <= 15) ? ki : ki + 33;
    Xcre[(size_t)q * 8192 + p] = lre[m * 65 + k];
    Xcim[(size_t)q * 8192 + p] = lim[m * 65 + k];
  }
}

// ---------------- xs[s,B,F] = sum_b wf[b,s] * X[B,F,b,m_s,k_s] ----------------
__global__ void k_gather_xs(const float* __restrict__ Xcre, const float* __restrict__ Xcim,
                            const float* __restrict__ wf,
                            const int* __restrict__ m_in, const int* __restrict__ k_in,
                            float* __restrict__ xsre, float* __restrict__ xsim) {
  int t = blockIdx.x * blockDim.x + threadIdx.x;
  if (t >= NSPEC * 128) return;
  int s = t >> 7;
  int r = t & 127;                 // Bb*32 + Fi
  int m64 = m_in[s], k64 = k_in[s];
  int mi = (m64 <= 15) ? m64 : m64 - 33;
  int ki = (k64 <= 15) ? k64 : k64 - 33;
  int q = mi * 31 + ki;
  size_t base = (size_t)q * 8192 + r * 64;   // 64 contiguous floats per lane
  const float4* xr4 = (const float4*)(Xcre + base);
  const float4* xi4 = (const float4*)(Xcim + base);
  float ar = 0.f, ai = 0.f;
  for (int b4 = 0; b4 < 16; ++b4) {
    float4 vr = xr4[b4];
    float4 vi = xi4[b4];
    int b = b4 << 2;
    float w0 = wf[(size_t)(b + 0) * NSPEC + s];
    float w1 = wf[(size_t)(b + 1) * NSPEC + s];
    float w2 = wf[(size_t)(b + 2) * NSPEC + s];
    float w3 = wf[(size_t)(b + 3) * NSPEC + s];
    ar += w0 * vr.x + w1 * vr.y + w2 * vr.z + w3 * vr.w;
    ai += w0 * vi.x + w1 * vi.y + w2 * vi.z + w3 * vi.w;
  }
  xsre[(size_t)s * 128 + r] = ar;
  xsim[(size_t)s * 128 + r] = ai;
}

// ---------------- y = scale * (D^T kernel) as a WMMA GEMM ----------------
// M = 5456 (s, 341 tiles), N = 2048 (r = i*64+o, 128 tiles), K = 72
// A[s][g] = D_t (re / im), B[g][r] = kern_t ; y[s*2048 + r]
__global__ void k_wmma_y(const float* __restrict__ Dre_t, const float* __restrict__ Dim_t,
                         const float* __restrict__ kern_t,
                         float* __restrict__ yre, float* __restrict__ yim) {
  int bid = blockIdx.x;
  int tm = bid >> 7;            // s-tile (0..340)
  int tn = bid & 127;           // r-tile (0..127)
  int lane = threadIdx.x;
  int lane15 = lane & 15;
  int khalf = (lane >> 4) << 1; // 0 or 2
  int srow = tm * 16 + lane15;  // < 5456 always (5456 = 341*16)
  int col  = tn * 16 + lane15;  // < 2048 always

  v8f accr = {0, 0, 0, 0, 0, 0, 0, 0};
  v8f acci = {0, 0, 0, 0, 0, 0, 0, 0};
  size_t abase = (size_t)srow * 72;

  for (int kk = 0; kk < 72; kk += 4) {
    int g0 = kk + khalf;
    v2f arr, aii, bb;
    arr.x = Dre_t[abase + g0]; arr.y = Dre_t[abase + g0 + 1];
    aii.x = Dim_t[abase + g0]; aii.y = Dim_t[abase + g0 + 1];
    size_t bbase = (size_t)g0 * 2048 + col;
    bb.x = kern_t[bbase]; bb.y = kern_t[bbase + 2048];
    accr = __builtin_amdgcn_wmma_f32_16x16x4_f32(false, arr, false, bb, (short)0, accr, false, false);
    acci = __builtin_amdgcn_wmma_f32_16x16x4_f32(false, aii, false, bb, (short)0, acci, false, false);
  }

  const float scale = 0.05892556509887896f; // 1/sqrt(288)
#pragma unroll
  for (int rr = 0; rr < 8; ++rr) {
    int row = tm * 16 + ((lane < 16) ? rr : rr + 8);
    size_t yi = (size_t)row * 2048 + col;
    yre[yi] = scale * accr[rr];
    yim[yi] = scale * acci[rr];
  }
}

// ---------------- block complex GEMM on WMMA: z = xb * conj(yb) ----------------
// per l: A[(m*4+b), (n*32+i)] (M=4L x K=32L), B[(n*32+i), (k*64+o)] (K x N=64L)
// Zr = Ar*Br + Ai*Bi ; Zi = Ai*Br - Ar*Bi   (conj folded in)
// Edge tiles: out-of-range lanes CLAMP their A-row / B-col address (each output
// row/col depends only on that lane's fragment, and those outputs are not stored).
__global__ void k_wmma_z(const float* __restrict__ xsre, const float* __restrict__ xsim,
                         const float* __restrict__ yre, const float* __restrict__ yim,
                         float* __restrict__ zre_t, float* __restrict__ zim_t) {
  int bid = blockIdx.x;
  int l = 0, tnc = 0, rem = bid;
  for (; l < 16; ++l) {
    int L = 2 * l + 1;
    int tmc = (4 * L + 15) >> 4;
    tnc = (64 * L + 15) >> 4;
    int tiles = tmc * tnc;
    if (rem < tiles) break;
    rem -= tiles;
  }
  if (l >= 16) return;
  int tm = rem / tnc, tn = rem % tnc;
  const int L = 2 * l + 1;
  const int off = l_off(l);
  const int M = 4 * L, Nn = 64 * L, K = 32 * L;

  int lane = threadIdx.x;           // one wave32 per block
  int lane15 = lane & 15;
  int khalf = (lane >> 4) << 1;     // K sub-offset 0 or 2 (f32 A/B fragment layout)
  int grow = tm * 16 + lane15;
  int gcol = tn * 16 + lane15;
  int growc = (grow < M) ? grow : (M - 1);   // clamp for edge tiles
  int gcolc = (gcol < Nn) ? gcol : (Nn - 1);
  int m_ = growc >> 2, b_ = growc & 3;
  int k_ = gcolc >> 6, o_ = gcolc & 63;

  size_t abase = (size_t)(off + m_ * L) * 128 + b_ * 32;
  size_t bbase = (size_t)(off + k_ * L) * 2048 + o_;

  v8f accr = {0, 0, 0, 0, 0, 0, 0, 0};
  v8f acci = {0, 0, 0, 0, 0, 0, 0, 0};

  for (int kk = 0; kk < K; kk += 4) {
    int c0 = kk + khalf;                     // even, so c0,c0+1 share the n row
    size_t a0 = abase + (size_t)((c0 >> 5) * 128 + (c0 & 31));
    v2f ar, ai;
    ar.x = xsre[a0]; ar.y = xsre[a0 + 1];
    ai.x = xsim[a0]; ai.y = xsim[a0 + 1];
    size_t b0 = bbase + (size_t)c0 * 64;     // B address is affine in c0
    v2f br, bi;
    br.x = yre[b0]; br.y = yre[b0 + 64];
    bi.x = yim[b0]; bi.y = yim[b0 + 64];
    v2f arn = -ar;  // f32 WMMA NEG bits only allow C-negate; negate A on VALU
    accr = __builtin_amdgcn_wmma_f32_16x16x4_f32(false, ar, false, br, (short)0, accr, false, false);
    accr = __builtin_amdgcn_wmma_f32_16x16x4_f32(false, ai, false, bi, (short)0, accr, false, false);
    acci = __builtin_amdgcn_wmma_f32_16x16x4_f32(false, ai, false, br, (short)0, acci, false, false);
    acci = __builtin_amdgcn_wmma_f32_16x16x4_f32(false, arn, false, bi, (short)0, acci, false, false);
  }

#pragma unroll
  for (int rr = 0; rr < 8; ++rr) {
    int lr = (lane < 16) ? rr : rr + 8;   // C/D fragment layout
    int orow = tm * 16 + lr;
    int ocol = tn * 16 + lane15;
    if (orow < M && ocol < Nn) {
      int mm = orow >> 2, bb = orow & 3;
      int kk2 = ocol >> 6, oo = ocol & 63;
      // transposed store: z_t[(b*64+o)][s] for unit-stride reads in k_gridf
      size_t zi = (size_t)(bb * 64 + oo) * NSPEC + (off + mm * L + kk2);
      zre_t[zi] = accr[rr];
      zim_t[zi] = acci[rr];
    }
  }
}

// ---------------- inverse Wigner + scatter (collision-free gather form) ----------------
// gridf[Bb,Fo,b,mh,kh] = sum_{l>=max(|m|,|k|)} wi[b, s(l,m,k)] * z[s, Bb, Fo]
__global__ void k_gridf(const float* __restrict__ zre_t, const float* __restrict__ zim_t,
                        const float* __restrict__ wiv,
                        float* __restrict__ gfre, float* __restrict__ gfim) {
  int t = blockIdx.x * blockDim.x + threadIdx.x;
  if (t >= 8388608) return;
  int kh = t & 31;
  int mh = (t >> 5) & 31;
  int b  = (t >> 10) & 31;
  int Fo = (t >> 15) & 63;
  int Bb = t >> 21;
  float ar = 0.f, ai = 0.f;
  if (mh != 16 && kh != 16) {
    int m = (mh < 16) ? mh : mh - 32;
    int k = (kh < 16) ? kh : kh - 32;
    int am = m < 0 ? -m : m;
    int ak = k < 0 ? -k : k;
    int lmin = am > ak ? am : ak;
    size_t zbase = (size_t)(Bb * 64 + Fo) * NSPEC;
    for (int l = lmin; l < 16; ++l) {
      int L = 2 * l + 1;
      int s = l_off(l) + (m + l) * L + (k + l);
      float w = wiv[(size_t)b * NSPEC + s];
      ar += w * zre_t[zbase + s];
      ai += w * zim_t[zbase + s];
    }
  }
  gfre[t] = ar;
  gfim[t] = ai;
}

// ---------------- inverse FFT (32x32), real part (n^2 cancels ifft norm) ----------------
__global__ void k_ifft(const float* __restrict__ gfre, const float* __restrict__ gfim,
                       float* __restrict__ out) {
  __shared__ float lre[32 * 33];
  __shared__ float lim[32 * 33];
  int p = blockIdx.x;
  int tid = threadIdx.x;
  size_t base = (size_t)p * 1024;
  for (int idx = tid; idx < 1024; idx += 256) {
    int u = idx >> 5, v = idx & 31;
    lre[u * 33 + v] = gfre[base + idx];
    lim[u * 33 + v] = gfim[base + idx];
  }
  __syncthreads();
  fft_pass<32>(lre, lim, 256, tid, false, 1.f);
  fft_pass<32>(lre, lim, 256, tid, true, 1.f);
  for (int idx = tid; idx < 1024; idx += 256) {
    int u = idx >> 5, v = idx & 31;
    out[base + idx] = lre[u * 33 + v];
  }
}

// ---------------- orchestration ----------------
extern "C" void kernel_launch(void* const* d_in, const int* in_sizes, int n_in,
                              void* d_out, int out_size, void* d_ws, size_t ws_size,
                              hipStream_t stream) {
  const float* x    = (const float*)d_in[0];   // (4,32,64,64,64)
  const float* kern = (const float*)d_in[1];   // (32,64,72)
  float* out = (float*)d_out;                  // (4,64,32,32,32)

  // workspace carve-up (fp32 units)
  float* p = (float*)d_ws;
  float* wf     = p; p += (size_t)64 * NSPEC;
  float* wiv    = p; p += (size_t)32 * NSPEC;
  float* Dre_t  = p; p += (size_t)NSPEC * 72;
  float* Dim_t  = p; p += (size_t)NSPEC * 72;
  float* kern_t = p; p += (size_t)72 * 2048;
  int* m_in = (int*)p; p += NSPEC;
  int* k_in = (int*)p; p += NSPEC;
  float* Xcre = p; p += (size_t)961 * 8192;
  float* Xcim = p; p += (size_t)961 * 8192;
  float* xsre = p; p += (size_t)NSPEC * 128;
  float* xsim = p; p += (size_t)NSPEC * 128;
  float* yre  = p; p += (size_t)NSPEC * 2048;
  float* yim  = p; p += (size_t)NSPEC * 2048;
  float* zre  = p; p += (size_t)NSPEC * 256;
  float* zim  = p; p += (size_t)NSPEC * 256;
  float* gfre = p; p += (size_t)8192 * 1024;
  float* gfim = p; p += (size_t)8192 * 1024;

  // constants
  k_wigner<<<dim3(64, 16), 64, 0, stream>>>(wf, 32, 1);   // forward, quadrature-weighted
  k_wigner<<<dim3(32, 16), 64, 0, stream>>>(wiv, 16, 0);  // inverse, (2l+1)-weighted
  k_dgrid<<<dim3(72, 16), 64, 0, stream>>>(Dre_t, Dim_t);
  k_indices<<<16, 64, 0, stream>>>(m_in, k_in);
  k_kern_t<<<(2048 * 72 + 255) / 256, 256, 0, stream>>>(kern, kern_t);

  // pipeline
  k_fft_fwd<<<8192, 256, 0, stream>>>(x, Xcre, Xcim);
  k_gather_xs<<<(NSPEC * 128 + 255) / 256, 256, 0, stream>>>(Xcre, Xcim, wf, m_in, k_in, xsre, xsim);

  k_wmma_y<<<341 * 128, 32, 0, stream>>>(Dre_t, Dim_t, kern_t, yre, yim);

  int tiles = 0;
  for (int l = 0; l < 16; ++l) {
    int L = 2 * l + 1;
    tiles += ((4 * L + 15) / 16) * ((64 * L + 15) / 16);
  }
  k_wmma_z<<<tiles, 32, 0, stream>>>(xsre, xsim, yre, yim, zre, zim);

  k_gridf<<<(8388608 + 255) / 256, 256, 0, stream>>>(zre, zim, wiv, gfre, gfim);
  k_ifft<<<8192, 256, 0, stream>>>(gfre, gfim, out);
}